// DSC_86071144612259
// MI455X (gfx1250) — compile-verified
//
#include <hip/hip_runtime.h>

typedef __attribute__((ext_vector_type(2))) float v2f;
typedef __attribute__((ext_vector_type(4))) float v4f;
typedef __attribute__((ext_vector_type(8))) float v8f;

#define BB    16384
#define PP    32
#define MCC   16
#define HH    24
#define MMM   64
#define NDEL  128          // delays 0..127 actually used
#define KTOT  4096         // NDEL * PP
#define ROWSTRIDE 4128     // 129 * 32 floats per y_rev row
#define KSPLIT 4
#define KRANGE (KTOT / KSPLIT)   // 1024 K per split
#define OUTN  (BB * MCC)         // 262144

// ---------------------------------------------------------------------------
// k0: W4s[i][l][n] = (lambda_e[i]*sigma[l])^0.25 * conv(phi_tilde[:,i], phi[:,l])[n]
//     n = 0..126  (maps to delay n+1)
// ---------------------------------------------------------------------------
__global__ void conv_scale_kernel(const float* __restrict__ phi,        // [64][24] (k,l)
                                  const float* __restrict__ phi_tilde,  // [64][24] (j,i)
                                  const float* __restrict__ sigma,      // [24]
                                  const float* __restrict__ lambda_e,   // [24]
                                  float* __restrict__ W4s) {            // [24][24][127]
  int t = blockIdx.x * blockDim.x + threadIdx.x;
  if (t >= HH * HH * 127) return;
  int n  = t % 127;
  int il = t / 127;
  int l  = il % HH;
  int i  = il / HH;
  int klo = (n - 63 > 0) ? n - 63 : 0;
  int khi = (n < 63) ? n : 63;
  float s = 0.0f;
  for (int k = klo; k <= khi; ++k)
    s += phi_tilde[k * HH + i] * phi[(n - k) * HH + l];
  float sc = sqrtf(sqrtf(lambda_e[i] * sigma[l]));   // inputs positive
  W4s[(i * HH + l) * 127 + n] = s * sc;
}

// ---------------------------------------------------------------------------
// k1: assemble transposed weight matrix Wt[(d*32+p)*16 + c]  (K-major, N=c inner)
// ---------------------------------------------------------------------------
__global__ void build_w_kernel(const float* __restrict__ M0,        // [16][32]
                               const float* __restrict__ M_tilde,   // [24][16][32]
                               const float* __restrict__ M_0l,      // [24][16][32]
                               const float* __restrict__ M_big,     // [24][24][16][32]
                               const float* __restrict__ sigma,
                               const float* __restrict__ lambda_e,
                               const float* __restrict__ phi,       // [64][24]
                               const float* __restrict__ phi_tilde, // [64][24]
                               const float* __restrict__ W4s,       // [576][127]
                               float* __restrict__ Wt) {            // [4096][16]
  int t = blockIdx.x * blockDim.x + threadIdx.x;   // [0, 128*16*32)
  int p = t & 31;
  int c = (t >> 5) & 15;
  int d = t >> 9;
  int cp = c * PP + p;

  float acc = (d == 0) ? M0[cp] : 0.0f;

  if (d >= 1 && d <= MMM) {              // Term 2, delay = j+1
    int j = d - 1;
    for (int i = 0; i < HH; ++i)
      acc += phi_tilde[j * HH + i] * sqrtf(sqrtf(lambda_e[i])) * M_tilde[i * (MCC * PP) + cp];
  }
  if (d <= MMM - 1) {                    // Term 3, delay = k
    for (int l = 0; l < HH; ++l)
      acc += phi[d * HH + l] * sqrtf(sqrtf(sigma[l])) * M_0l[l * (MCC * PP) + cp];
  }
  if (d >= 1) {                          // Term 4, delay = n+1
    int n = d - 1;
    for (int il = 0; il < HH * HH; ++il)
      acc += W4s[il * 127 + n] * M_big[il * (MCC * PP) + cp];
  }
  Wt[(d * PP + p) * MCC + c] = acc;
}

// ---------------------------------------------------------------------------
// k2: split-K GEMM partials. grid = (BB/128, KSPLIT), block = 256 (8 waves).
// Wave w owns batch rows [blk.x*128 + w*16, +16), K range [blk.y*KRANGE, +KRANGE).
// part[blk.y][b][c] accumulated via V_WMMA_F32_16X16X4_F32.
// ---------------------------------------------------------------------------
__global__ void __launch_bounds__(256) gemm_kernel(const float* __restrict__ Y,
                                                   const float* __restrict__ Wt,
                                                   float* __restrict__ part) {
  __shared__ float lds[256 * MCC];                 // 16 KB chunk of Wt

  const int tid  = threadIdx.x;
  const int wave = tid >> 5;
  const int lane = tid & 31;
  const int ln15 = lane & 15;                      // A row within tile / D column
  const int hi   = lane >> 4;                      // 0: K={0,1}, 1: K={2,3}
  const int koff = hi << 1;

  const long tileBase = (long)blockIdx.x * 128 + (long)wave * 16;
  const int  kbase    = blockIdx.y * KRANGE;
  const float* yrow = Y + (tileBase + ln15) * (long)ROWSTRIDE;

  v8f acc = {};

  for (int kc = kbase; kc < kbase + KRANGE; kc += 256) {
    __syncthreads();
    // cooperative stage: thread t copies Wt row (kc+t): 16 floats = 4x float4
    const v4f* src = (const v4f*)(Wt + (long)(kc + tid) * MCC);
    v4f* dst = (v4f*)(lds + tid * MCC);
    dst[0] = src[0]; dst[1] = src[1]; dst[2] = src[2]; dst[3] = src[3];
    __syncthreads();

#pragma unroll 4
    for (int kk = 0; kk < 256; kk += 4) {
      // A fragment: lanes 0-15 read K={k0,k0+1}, lanes 16-31 read K={k0+2,k0+3}
      v2f a = *(const v2f*)(yrow + kc + kk + koff);
      // B fragment from LDS: Wt[k0+koff][n], Wt[k0+koff+1][n], n = ln15
      v2f b;
      b.x = lds[(kk + koff) * MCC + ln15];
      b.y = lds[(kk + koff + 1) * MCC + ln15];
      acc = __builtin_amdgcn_wmma_f32_16x16x4_f32(
          /*neg_a=*/false, a, /*neg_b=*/false, b,
          /*c_mod=*/(short)0, acc, /*reuse_a=*/false, /*reuse_b=*/false);
    }
  }

  // D layout: VGPR r, lanes 0-15 -> M=r, lanes 16-31 -> M=r+8; N = ln15
  float* o = part + (long)blockIdx.y * OUTN + (tileBase + (long)hi * 8) * MCC + ln15;
#pragma unroll
  for (int r = 0; r < 8; ++r)
    o[(long)r * MCC] = acc[r];
}

// ---------------------------------------------------------------------------
// k3: fixed-order split-K reduction (deterministic)
// ---------------------------------------------------------------------------
__global__ void reduce_kernel(const float* __restrict__ part, float* __restrict__ out) {
  int t = blockIdx.x * blockDim.x + threadIdx.x;   // [0, OUTN)
  float s = part[t];
#pragma unroll
  for (int sp = 1; sp < KSPLIT; ++sp)
    s += part[(long)sp * OUTN + t];
  out[t] = s;
}

// ---------------------------------------------------------------------------
extern "C" void kernel_launch(void* const* d_in, const int* in_sizes, int n_in,
                              void* d_out, int out_size, void* d_ws, size_t ws_size,
                              hipStream_t stream) {
  const float* y_rev     = (const float*)d_in[0];
  const float* M0        = (const float*)d_in[1];
  const float* M_tilde   = (const float*)d_in[2];
  const float* M_0l      = (const float*)d_in[3];
  const float* M_big     = (const float*)d_in[4];
  const float* sigma     = (const float*)d_in[5];
  const float* lambda_e  = (const float*)d_in[6];
  const float* phi       = (const float*)d_in[7];
  const float* phi_tilde = (const float*)d_in[8];
  float* out = (float*)d_out;

  // workspace layout (floats): W4s @ 0 (73152), Wt @ 80000 (65536),
  // part @ 160000 (KSPLIT*OUTN = 1048576).  Total ~4.84 MB.
  float* W4s  = (float*)d_ws;
  float* Wt   = (float*)d_ws + 80000;
  float* part = (float*)d_ws + 160000;

  conv_scale_kernel<<<(HH * HH * 127 + 255) / 256, 256, 0, stream>>>(
      phi, phi_tilde, sigma, lambda_e, W4s);

  build_w_kernel<<<(NDEL * MCC * PP) / 256, 256, 0, stream>>>(
      M0, M_tilde, M_0l, M_big, sigma, lambda_e, phi, phi_tilde, W4s, Wt);

  dim3 grid(BB / 128, KSPLIT);
  gemm_kernel<<<grid, 256, 0, stream>>>(y_rev, Wt, part);

  reduce_kernel<<<OUTN / 256, 256, 0, stream>>>(part, out);
}